// ImputerLoss_90709709291684
// MI455X (gfx1250) — compile-verified
//
#include <hip/hip_runtime.h>
#include <hip/hip_bf16.h>
#include <stdint.h>

// Problem-shape caps (reference: T=2048, N=64, C=128, S=256 -> L=513)
#define MAX_S 256
#define MAX_L (2 * MAX_S + 1) /* 513 */
#define MAX_T 2048
#define MAX_C 128
#define NEGF (-1e30f)

// ---- CDNA5 async global->LDS row DMA (ASYNCcnt-tracked), inline asm so it
// works on both ROCm7.2 and amdgpu-toolchain. GVS mode: 64-bit SGPR base +
// 32-bit per-lane byte offset; VDST VGPR carries the LDS byte address.
__device__ __forceinline__ void async_row_load_b128(uint32_t lds_addr,
                                                    uint32_t byte_off,
                                                    const float* gbase) {
  asm volatile("global_load_async_to_lds_b128 %0, %1, %2"
               :: "v"(lds_addr), "v"(byte_off), "s"(gbase)
               : "memory");
}

__device__ __forceinline__ float lse3(float a0, float a1, float a2) {
  float m = fmaxf(a0, fmaxf(a1, a2));
  return m + __logf(__expf(a0 - m) + __expf(a1 - m) + __expf(a2 - m));
}

__global__ void __launch_bounds__(512)
imputer_alpha_kernel(const float* __restrict__ log_prob,   // (T,N,C)
                     const int* __restrict__ targets,      // (N,S)
                     const int* __restrict__ force_emits,  // (N,T)
                     const int* __restrict__ input_lengths,
                     const int* __restrict__ target_lengths,
                     float* __restrict__ loss_out,         // (N,)
                     int T, int N, int C, int S) {
  __shared__ float alpha[2][MAX_L + 3];
  __shared__ __align__(16) float rowbuf[3][MAX_C]; // triple-buffered emit rows
  __shared__ int extp[MAX_L + 3];                  // ext label | (skip<<8)
  __shared__ int feld[MAX_T];                      // force_emits[n, :]

  const int n = blockIdx.x;
  const int tid = threadIdx.x;
  const int L = 2 * S + 1;

  // Extended labels + skip flags (skip only at odd s>=3 with differing labels)
  for (int s = tid; s < L; s += blockDim.x) {
    int e = 0, sk = 0;
    if (s & 1) {
      int j = s >> 1;
      e = targets[n * S + j];
      if (s >= 3) sk = (e != targets[n * S + j - 1]) ? 1 : 0;
    }
    extp[s] = e | (sk << 8);
  }
  for (int t = tid; t < T; t += blockDim.x)
    feld[t] = force_emits[n * T + t];

  // Low 32 bits of a generic pointer to LDS == LDS byte address on gfx1250.
  const uint32_t lds_row0 = (uint32_t)(uintptr_t)(&rowbuf[0][0]);

  // Prologue: issue rows 0..2, wait for row 0 (in-order completion).
  if (tid < 32) {
    uint32_t voff = (uint32_t)tid * 16u; // 32 lanes * 16B = 512B = 128 floats
    for (int r = 0; r < 3; ++r) {
      if (r < T)
        async_row_load_b128(lds_row0 + (uint32_t)r * (MAX_C * 4) + voff, voff,
                            log_prob + ((size_t)r * N + n) * C);
    }
    asm volatile("s_wait_asynccnt 2" ::: "memory");
  }
  __syncthreads();

  const int inlen = input_lengths[n];

  // t = 0 init: alpha = (s<=1 ? emit : NEG), then force mask.
  {
    const int fe0 = feld[0];
    for (int s = tid; s < L; s += blockDim.x) {
      float emit = rowbuf[0][extp[s] & 0xFF];
      float a = (s <= 1) ? emit : NEGF;
      if (!(fe0 < 0 || s == fe0)) a = NEGF;
      alpha[0][s] = a;
    }
  }

  int cur = 0;
  for (int t = 1; t < T; ++t) {
    // Outstanding async loads here are a subset of {row t, row t+1}.
    if (tid < 32) {
      if (t + 1 < T) asm volatile("s_wait_asynccnt 1" ::: "memory");
      else           asm volatile("s_wait_asynccnt 0" ::: "memory");
    }
    __syncthreads(); // retires compute(t-1); row t visible to all waves

    // Prefetch row t+2 into buffer (t+2)%3 == (t-1)%3, now safely reusable.
    if (tid < 32 && (t + 2) < T) {
      uint32_t voff = (uint32_t)tid * 16u;
      int buf = (t + 2) % 3;
      async_row_load_b128(lds_row0 + (uint32_t)buf * (MAX_C * 4) + voff, voff,
                          log_prob + ((size_t)(t + 2) * N + n) * C);
    }

    const int fe = feld[t];
    const float* rb = rowbuf[t % 3];
    const float* ac = alpha[cur];
    float* an = alpha[cur ^ 1];
    const bool active = (t < inlen);

    for (int s = tid; s < L; s += blockDim.x) {
      int p = extp[s];
      float a0 = ac[s];
      float a1 = (s >= 1) ? ac[s - 1] : NEGF;
      float a2 = (p & 0x100) ? ac[s - 2] : NEGF;
      float nw = rb[p & 0xFF] + lse3(a0, a1, a2);
      if (!(fe < 0 || s == fe)) nw = NEGF;
      an[s] = active ? nw : a0;
    }
    cur ^= 1;
  }
  __syncthreads();

  if (tid == 0) {
    int tl = target_lengths[n];
    int hi = 2 * tl; if (hi > L - 1) hi = L - 1;
    int lo = hi - 1; if (lo < 0) lo = 0;
    float ahi = alpha[cur][hi];
    float ll;
    if (tl > 0) {
      float alo = alpha[cur][lo];
      float m = fmaxf(ahi, alo);
      ll = m + __logf(__expf(ahi - m) + __expf(alo - m));
    } else {
      ll = ahi;
    }
    float tlf = (float)(tl > 0 ? tl : 1);
    loss_out[n] = -ll / tlf;
  }
}

// ---- Final mean over N=64 losses via V_WMMA_F32_16X16X4_F32 (f32 in/out).
// B(4x16) holds the losses (B layout: vgpr j, lanes<16 -> K=j; lanes>=16 ->
// K=j+2; N = lane%16), A = ones -> D(m,n) = column sum of 4 losses, identical
// across m; reduce the 16 column sums and scale by 1/64.
typedef __attribute__((ext_vector_type(2))) float v2f;
typedef __attribute__((ext_vector_type(8))) float v8f;

__global__ void reduce_mean_kernel(const float* __restrict__ losses,
                                   float* __restrict__ out, int n) {
  const int lane = threadIdx.x; // launched with exactly 32 threads (EXEC all 1s)
#if defined(__gfx1250__) && __has_builtin(__builtin_amdgcn_wmma_f32_16x16x4_f32)
  if (n == 64) {
    int col = lane & 15;
    int kb = (lane >> 4) * 2;
    v2f a; a.x = 1.0f; a.y = 1.0f;
    v2f b; b.x = losses[col * 4 + kb + 0];
    b.y = losses[col * 4 + kb + 1];
    v8f c = {};
    v8f d = __builtin_amdgcn_wmma_f32_16x16x4_f32(
        false, a, false, b, (short)0, c, false, false);
    __shared__ float cs[16];
    if (lane < 16) cs[lane] = d[0]; // D(0, col) = colsum[col]
    __syncthreads();
    if (lane == 0) {
      float ssum = 0.f;
      for (int i = 0; i < 16; ++i) ssum += cs[i];
      out[0] = ssum * (1.0f / 64.0f);
    }
    return;
  }
#endif
  if (lane == 0) {
    float ssum = 0.f;
    for (int i = 0; i < n; ++i) ssum += losses[i];
    out[0] = ssum / (float)n;
  }
}

extern "C" void kernel_launch(void* const* d_in, const int* in_sizes, int n_in,
                              void* d_out, int out_size, void* d_ws,
                              size_t ws_size, hipStream_t stream) {
  (void)n_in; (void)out_size; (void)ws_size;
  const float* log_prob       = (const float*)d_in[0];
  const int*   targets        = (const int*)d_in[1];
  const int*   force_emits    = (const int*)d_in[2];
  const int*   input_lengths  = (const int*)d_in[3];
  const int*   target_lengths = (const int*)d_in[4];
  float* out = (float*)d_out;

  const int N = in_sizes[3];
  const int T = in_sizes[2] / N;
  const int S = in_sizes[1] / N;
  const int C = in_sizes[0] / (T * N);

  float* losses = (float*)d_ws; // N floats of scratch

  imputer_alpha_kernel<<<N, 512, 0, stream>>>(
      log_prob, targets, force_emits, input_lengths, target_lengths, losses,
      T, N, C, S);
  reduce_mean_kernel<<<1, 32, 0, stream>>>(losses, out, N);
}